// FTL_27032524161043
// MI455X (gfx1250) — compile-verified
//
#include <hip/hip_runtime.h>

typedef __attribute__((ext_vector_type(2))) float v2f;
typedef __attribute__((ext_vector_type(8))) float v8f;

// out[b, g*3+i, p] = sum_j P[b,i,j] * z[b, g*4+j, p]
//   z: (16, 512, 64, 64) f32, P: (16, 3, 4) f32, out: (16, 384, 64, 64) f32
//
// One V_WMMA_F32_16X16X4_F32 per 16-position chunk:
//   A (16x4)  = P[b] zero-padded to 16 rows (loop-invariant per wave)
//   B (4x16)  = z channels j=0..3 at 16 consecutive spatial positions
//   D (16x16) = rows 0..2 (VGPRs 0..2, lanes 0-15) are the 3 output planes
//
// ISA VGPR layouts (CDNA5 ISA):
//   A 16x4 f32: VGPR0 = K=(lane<16?0:2), VGPR1 = K+1, row = lane&15
//   B 4x16 f32: VGPR0 = row K=(lane<16?0:2), VGPR1 = K+1, col = lane&15
//   D 16x16 f32: VGPR r, lanes 0-15 => M=r, N=lane
//
// Shapes are compile-time so every load/store in the unrolled group folds its
// chunk offset into the 24-bit IOFFSET immediate (no per-iteration 64-bit
// address arithmetic), and the compiler can clause the NT loads/stores.
__global__ __launch_bounds__(256) void ftl_proj_wmma(
    const float* __restrict__ z, const float* __restrict__ pm,
    float* __restrict__ out) {
  constexpr int HW = 64 * 64;   // spatial plane
  constexpr int G  = 128;       // channel groups (512 / 4)
  constexpr int NCHUNK = HW / 16;

  const int b    = blockIdx.y;
  const int g    = blockIdx.x;
  const int lane = threadIdx.x & 31;
  const int wave = threadIdx.x >> 5;      // 0..7
  const int col  = lane & 15;             // N (position in chunk) / M (row of A)
  const int k0   = (lane >> 4) << 1;      // K pair: 0 for lanes 0-15, 2 for 16-31

  // ---- A: projection matrix, rows >= 3 are zero
  const float* p = pm + b * 12;           // P[b] is 3x4 row-major
  v2f a;
  a.x = (col < 3) ? p[col * 4 + k0]     : 0.0f;
  a.y = (col < 3) ? p[col * 4 + k0 + 1] : 0.0f;

  const size_t zbase = ((size_t)b * (G * 4) + (size_t)g * 4) * (size_t)HW;
  const size_t obase = ((size_t)b * (G * 3) + (size_t)g * 3) * (size_t)HW;
  const float* zp0 = z + zbase + (size_t)k0 * HW + col;  // channel k0
  const float* zp1 = zp0 + HW;                           // channel k0+1
  float* o0 = out + obase + col;

#pragma unroll 8
  for (int c = wave; c < NCHUNK; c += 8) {
    const int pos = c << 4;
    // prefetch this wave's chunk one stride ahead (streaming, speculative)
    __builtin_prefetch(zp0 + pos + (8 << 4), 0, 0);

    v2f bm;                               // B matrix: z[j][pos+col]
    bm.x = __builtin_nontemporal_load(zp0 + pos);
    bm.y = __builtin_nontemporal_load(zp1 + pos);

    v8f acc = {};
    acc = __builtin_amdgcn_wmma_f32_16x16x4_f32(
        /*neg_a=*/false, a, /*neg_b=*/false, bm,
        /*c_mod=*/(short)0, acc, /*reuse_a=*/false, /*reuse_b=*/false);

    if (lane < 16) {                      // rows 0..2 of D live in lanes 0-15
      __builtin_nontemporal_store(acc[0], o0 + 0 * HW + pos);
      __builtin_nontemporal_store(acc[1], o0 + 1 * HW + pos);
      __builtin_nontemporal_store(acc[2], o0 + 2 * HW + pos);
    }
  }
}

extern "C" void kernel_launch(void* const* d_in, const int* in_sizes, int n_in,
                              void* d_out, int out_size, void* d_ws, size_t ws_size,
                              hipStream_t stream) {
  (void)in_sizes; (void)n_in; (void)d_ws; (void)ws_size; (void)out_size;
  const float* z  = (const float*)d_in[0];   // (16, 512, 64, 64) f32
  const float* pm = (const float*)d_in[1];   // (16, 3, 4) f32
  float* out = (float*)d_out;                // (16, 384, 64, 64) f32

  dim3 grid(128, 16);                        // (g, b)
  ftl_proj_wmma<<<grid, 256, 0, stream>>>(z, pm, out);
}